// SGCN_9758165697214
// MI455X (gfx1250) — compile-verified
//
#include <hip/hip_runtime.h>
#include <hip/hip_bf16.h>
#include <math.h>

typedef __attribute__((ext_vector_type(2))) float v2f;
typedef __attribute__((ext_vector_type(8))) float v8f;

#define WMMA_F32(a, b, c) \
  __builtin_amdgcn_wmma_f32_16x16x4_f32(false, (a), false, (b), (short)0, (c), false, false)

// ---------------------------------------------------------------- utilities
__global__ void k_zero_i32(int* __restrict__ p, int n) {
  int i = blockIdx.x * blockDim.x + threadIdx.x;
  if (i < n) p[i] = 0;
}
__global__ void k_zero_f32(float* __restrict__ p, int n) {
  int i = blockIdx.x * blockDim.x + threadIdx.x;
  if (i < n) p[i] = 0.0f;
}

// ---------------------------------------------------------------- CSR build
__global__ void k_count(const int* __restrict__ dst, int* __restrict__ cnt, int e) {
  int i = blockIdx.x * blockDim.x + threadIdx.x;
  if (i < e) atomicAdd(&cnt[dst[i]], 1);
}

// deg = (#in-edges) + 1 (self loop); dinv = deg^-0.5  (deg >= 1 always)
__global__ void k_dinv(const int* __restrict__ cnt, float* __restrict__ dinv, int n) {
  int i = blockIdx.x * blockDim.x + threadIdx.x;
  if (i < n) dinv[i] = 1.0f / sqrtf((float)(cnt[i] + 1));
}

__global__ void k_scan_block(const int* __restrict__ cnt, int* __restrict__ out,
                             int* __restrict__ bsum, int n) {
  __shared__ int sh[256];
  int i = blockIdx.x * 256 + threadIdx.x;
  int v = (i < n) ? cnt[i] : 0;
  sh[threadIdx.x] = v;
  __syncthreads();
  for (int off = 1; off < 256; off <<= 1) {
    int t = 0;
    if ((int)threadIdx.x >= off) t = sh[threadIdx.x - off];
    __syncthreads();
    if ((int)threadIdx.x >= off) sh[threadIdx.x] += t;
    __syncthreads();
  }
  if (i < n) out[i] = sh[threadIdx.x] - v;   // exclusive
  if (threadIdx.x == 255) bsum[blockIdx.x] = sh[255];
}

__global__ void k_scan_top(int* __restrict__ bsum, int nb) {
  __shared__ int sh[512];
  int v = ((int)threadIdx.x < nb) ? bsum[threadIdx.x] : 0;
  sh[threadIdx.x] = v;
  __syncthreads();
  for (int off = 1; off < 512; off <<= 1) {
    int t = 0;
    if ((int)threadIdx.x >= off) t = sh[threadIdx.x - off];
    __syncthreads();
    if ((int)threadIdx.x >= off) sh[threadIdx.x] += t;
    __syncthreads();
  }
  if ((int)threadIdx.x < nb) bsum[threadIdx.x] = sh[threadIdx.x] - v;  // exclusive
}

__global__ void k_scan_add(int* __restrict__ rowptr, const int* __restrict__ bsum,
                           int n, int e) {
  int i = blockIdx.x * 256 + threadIdx.x;
  if (i < n) rowptr[i] += bsum[blockIdx.x];
  if (i == 0) rowptr[n] = e;
}

__global__ void k_fill(const int* __restrict__ src, const int* __restrict__ dst,
                       const float* __restrict__ ea, const float* __restrict__ dinv,
                       const int* __restrict__ rowptr, int* __restrict__ cursor,
                       int* __restrict__ csrc, float* __restrict__ cw, int e) {
  int i = blockIdx.x * blockDim.x + threadIdx.x;
  if (i >= e) return;
  int d = dst[i], s = src[i];
  int pos = rowptr[d] + atomicAdd(&cursor[d], 1);
  csrc[pos] = s;
  cw[pos] = dinv[s] * dinv[d] * expf(-ea[i]);
}

// ---------------------------------------------------------------- GEMM via fp32 WMMA
// T[n,64] = X[n,K] @ W[K,64]. One wave computes a 16x64 output strip.
// A frag (16x4 f32): lane m=lane&15, holds X[m][ka], X[m][ka+1], ka = k0 + 2*(lane>>4)
// B frag (4x16 f32): lane n=lane&15, holds W[ka][n], W[ka+1][n]
// C/D (16x16 f32): vgpr r, lanes0-15 -> M=r, lanes16-31 -> M=r+8, N=lane&15
template <int K>
__global__ __launch_bounds__(256) void k_gemm_wmma(const float* __restrict__ X,
                                                   const float* __restrict__ W,
                                                   float* __restrict__ T, int nrows) {
  int wv = blockIdx.x * 8 + ((int)threadIdx.x >> 5);
  int lane = threadIdx.x & 31;
  int row_base = wv * 16;
  if (row_base >= nrows) return;            // wave-uniform: EXEC stays all-ones
  int m = lane & 15;
  int khalf = (lane >> 4) << 1;             // 0 or 2
  const float* xrow = X + (size_t)(row_base + m) * K;
  v8f c0 = {}, c1 = {}, c2 = {}, c3 = {};
#pragma unroll 4
  for (int k0 = 0; k0 < K; k0 += 4) {
    int ka = k0 + khalf;
    v2f a;
    a.x = xrow[ka];
    a.y = xrow[ka + 1];
    const float* wp = W + (size_t)ka * 64 + m;   // wp[64] is row ka+1
    v2f b0; b0.x = wp[0];  b0.y = wp[64];
    v2f b1; b1.x = wp[16]; b1.y = wp[80];
    v2f b2; b2.x = wp[32]; b2.y = wp[96];
    v2f b3; b3.x = wp[48]; b3.y = wp[112];
    c0 = WMMA_F32(a, b0, c0);
    c1 = WMMA_F32(a, b1, c1);
    c2 = WMMA_F32(a, b2, c2);
    c3 = WMMA_F32(a, b3, c3);
  }
  int mh = (lane >> 4) * 8;
  float* out = T + (size_t)(row_base + mh) * 64 + m;
#pragma unroll
  for (int r = 0; r < 8; ++r) {
    float* o = out + (size_t)r * 64;
    o[0]  = c0[r];
    o[16] = c1[r];
    o[32] = c2[r];
    o[48] = c3[r];
  }
}

// ---------------------------------------------------------------- SpMM (pull) + bias + ReLU
__global__ __launch_bounds__(256) void k_agg(const float* __restrict__ T,
                                             const int* __restrict__ rowptr,
                                             const int* __restrict__ csrc,
                                             const float* __restrict__ cw,
                                             const float* __restrict__ dinv,
                                             const float* __restrict__ bias,
                                             float* __restrict__ H, int n) {
  int node = blockIdx.x * 4 + ((int)threadIdx.x >> 6);
  if (node >= n) return;
  int f = threadIdx.x & 63;
  float di = dinv[node];
  float acc = di * di * T[(size_t)node * 64 + f];   // self loop: exp(-0)=1
  int e0 = rowptr[node], e1 = rowptr[node + 1];
  for (int e = e0; e < e1; ++e) {
    acc = fmaf(cw[e], T[(size_t)csrc[e] * 64 + f], acc);
  }
  acc += bias[f];
  H[(size_t)node * 64 + f] = fmaxf(acc, 0.0f);
}

// ---------------------------------------------------------------- pooling
__global__ void k_pool_count(const int* __restrict__ batch, float* __restrict__ counts,
                             int n, int ngraphs) {
  int g = threadIdx.x;
  if (g >= ngraphs) return;
  // lower_bound(g) and lower_bound(g+1) on sorted batch
  int lo = 0, hi = n;
  while (lo < hi) { int mid = (lo + hi) >> 1; if (batch[mid] < g) lo = mid + 1; else hi = mid; }
  int a = lo;
  lo = 0; hi = n;
  while (lo < hi) { int mid = (lo + hi) >> 1; if (batch[mid] < g + 1) lo = mid + 1; else hi = mid; }
  counts[g] = (float)(lo - a);
}

__global__ __launch_bounds__(64) void k_pool_sum(const float* __restrict__ H,
                                                 const int* __restrict__ batch,
                                                 float* __restrict__ pooled, int n) {
  int f = threadIdx.x;  // 64 features
  int per = (n + gridDim.x - 1) / gridDim.x;
  int beg = blockIdx.x * per;
  int end = beg + per; if (end > n) end = n;
  if (beg >= end) return;
  int curg = batch[beg];
  float acc = 0.0f;
  for (int node = beg; node < end; ++node) {
    int g = batch[node];
    if (g != curg) {
      atomicAdd(&pooled[(size_t)curg * 64 + f], acc);
      acc = 0.0f;
      curg = g;
    }
    acc += H[(size_t)node * 64 + f];
  }
  atomicAdd(&pooled[(size_t)curg * 64 + f], acc);
}

__global__ void k_classify(const float* __restrict__ pooled, const float* __restrict__ counts,
                           const float* __restrict__ Wc, const float* __restrict__ bc,
                           float* __restrict__ out, int ngraphs, int nclasses) {
  int idx = blockIdx.x * blockDim.x + threadIdx.x;
  if (idx >= ngraphs * nclasses) return;
  int g = idx / nclasses;
  int c = idx - g * nclasses;
  float inv = 1.0f / fmaxf(counts[g], 1.0f);
  float acc = bc[c];
#pragma unroll 8
  for (int h = 0; h < 64; ++h)
    acc = fmaf(pooled[(size_t)g * 64 + h] * inv, Wc[(size_t)h * nclasses + c], acc);
  out[idx] = acc;
}

// ---------------------------------------------------------------- host launch
extern "C" void kernel_launch(void* const* d_in, const int* in_sizes, int n_in,
                              void* d_out, int out_size, void* d_ws, size_t ws_size,
                              hipStream_t stream) {
  const float* x        = (const float*)d_in[0];
  const int*   edge_src = (const int*)d_in[1];
  const int*   edge_dst = (const int*)d_in[2];
  const float* edge_ea  = (const float*)d_in[3];
  const int*   batch    = (const int*)d_in[4];
  const float* W0 = (const float*)d_in[5];
  const float* b0 = (const float*)d_in[6];
  const float* W1 = (const float*)d_in[7];
  const float* b1 = (const float*)d_in[8];
  const float* W2 = (const float*)d_in[9];
  const float* b2 = (const float*)d_in[10];
  const float* Wc = (const float*)d_in[11];
  const float* bc = (const float*)d_in[12];
  float* out = (float*)d_out;

  const int N   = in_sizes[4];            // 100000
  const int E   = in_sizes[1];            // 3200000
  const int FIN = in_sizes[0] / N;        // 128
  const int C   = in_sizes[12];           // 196
  const int B   = out_size / C;           // 16

  // workspace carve-up (256B aligned)
  char* base = (char*)d_ws;
  size_t off = 0;
  auto carve = [&](size_t bytes) -> char* {
    char* p = base + off;
    off = (off + bytes + 255) & ~(size_t)255;
    return p;
  };
  int*   cnt    = (int*)  carve((size_t)N * 4);
  int*   rowptr = (int*)  carve((size_t)(N + 1) * 4);
  int*   cursor = (int*)  carve((size_t)N * 4);
  float* dinv   = (float*)carve((size_t)N * 4);
  int*   bsum   = (int*)  carve(512 * 4);
  float* pooled = (float*)carve((size_t)B * 64 * 4);
  float* counts = (float*)carve((size_t)B * 4);
  int*   csrc   = (int*)  carve((size_t)E * 4);
  float* cw     = (float*)carve((size_t)E * 4);
  float* Tbuf   = (float*)carve((size_t)N * 64 * 4);
  float* Hbuf   = (float*)carve((size_t)N * 64 * 4);
  (void)ws_size; (void)n_in;

  const int nbN = (N + 255) / 256;
  const int nbE = (E + 255) / 256;

  // ---- build degree / dinv / CSR (recomputed every call: deterministic work)
  k_zero_i32<<<nbN, 256, 0, stream>>>(cnt, N);
  k_count<<<nbE, 256, 0, stream>>>(edge_dst, cnt, E);
  k_dinv<<<nbN, 256, 0, stream>>>(cnt, dinv, N);
  k_scan_block<<<nbN, 256, 0, stream>>>(cnt, rowptr, bsum, N);
  k_scan_top<<<1, 512, 0, stream>>>(bsum, nbN);
  k_scan_add<<<nbN, 256, 0, stream>>>(rowptr, bsum, N, E);
  k_zero_i32<<<nbN, 256, 0, stream>>>(cursor, N);
  k_fill<<<nbE, 256, 0, stream>>>(edge_src, edge_dst, edge_ea, dinv, rowptr, cursor,
                                  csrc, cw, E);

  // ---- 3 SGCN layers: WMMA GEMM then CSR pull-aggregate (+bias, ReLU)
  const int gemmGrid = (N + 127) / 128;   // 8 waves/block, 16 rows/wave
  const int aggGrid  = (N + 3) / 4;

  if (FIN == 128)
    k_gemm_wmma<128><<<gemmGrid, 256, 0, stream>>>(x, W0, Tbuf, N);
  else
    k_gemm_wmma<64><<<gemmGrid, 256, 0, stream>>>(x, W0, Tbuf, N);
  k_agg<<<aggGrid, 256, 0, stream>>>(Tbuf, rowptr, csrc, cw, dinv, b0, Hbuf, N);

  k_gemm_wmma<64><<<gemmGrid, 256, 0, stream>>>(Hbuf, W1, Tbuf, N);
  k_agg<<<aggGrid, 256, 0, stream>>>(Tbuf, rowptr, csrc, cw, dinv, b1, Hbuf, N);

  k_gemm_wmma<64><<<gemmGrid, 256, 0, stream>>>(Hbuf, W2, Tbuf, N);
  k_agg<<<aggGrid, 256, 0, stream>>>(Tbuf, rowptr, csrc, cw, dinv, b2, Hbuf, N);

  // ---- global mean pool + classifier
  k_zero_f32<<<(B * 64 + 255) / 256, 256, 0, stream>>>(pooled, B * 64);
  k_pool_count<<<1, 32, 0, stream>>>(batch, counts, N, B);
  k_pool_sum<<<256, 64, 0, stream>>>(Hbuf, batch, pooled, N);
  k_classify<<<(B * C + 255) / 256, 256, 0, stream>>>(pooled, counts, Wc, bc, out, B, C);
}